// mambaflow_head_8100308320796
// MI455X (gfx1250) — compile-verified
//
#include <hip/hip_runtime.h>
#include <math.h>

typedef float v2f __attribute__((ext_vector_type(2)));
typedef float v8f __attribute__((ext_vector_type(8)));

#define GRID3 262144          // 64^3
#define CVOX 96
#define CPT 32
#define OFFD 64
#define DEC 192               // K of first GEMM
#define HID 96                // output channels of first GEMM
#define W1T_STRIDE 196        // LDS stride for transposed W1 columns (bank-conflict free b64)
#define FEAT_STRIDE 196       // padded LDS stride for feature rows
#define H_STRIDE 97           // padded LDS stride for h rows (reuses feat region)

// Dynamic LDS layout (floats)
#define W1T_FLOATS   (HID * W1T_STRIDE)       // 18816  (W1 stored transposed: [n][k])
#define FEAT_FLOATS  (8 * 16 * FEAT_STRIDE)   // 25088
#define SMEM_FLOATS  (W1T_FLOATS + FEAT_FLOATS + HID + HID*3 + 4 + 192 + OFFD) // 44548

__global__ __launch_bounds__(256) void mambaflow_head_kernel(
    const float* __restrict__ vox,    // (96,64,64,64)
    const int*   __restrict__ coords, // (N,3) x,y,z
    const float* __restrict__ pfeat,  // (N,32)
    const float* __restrict__ poff,   // (N,3)
    const float* __restrict__ Woff,   // (3,64)
    const float* __restrict__ boff,   // (64)
    const float* __restrict__ W1,     // (192,96)
    const float* __restrict__ b1,     // (96)
    const float* __restrict__ W2,     // (96,3)
    const float* __restrict__ b2,     // (3)
    float* __restrict__ out,          // (N,3)
    int npts)
{
    extern __shared__ float smem[];
    float* W1Ts  = smem;                         // 96*196, transposed: W1Ts[n*196 + k]
    float* feats = W1Ts + W1T_FLOATS;            // 8 waves * 16 * 196
    float* b1s   = feats + FEAT_FLOATS;          // 96
    float* W2s   = b1s + HID;                    // 288
    float* b2s   = W2s + HID * 3;                // 4
    float* Woffs = b2s + 4;                      // 192
    float* boffs = Woffs + 192;                  // 64

    const int tid = threadIdx.x;

    // ---- Stage weights into LDS (block-cooperative); W1 is transposed on the fly ----
    for (int i = tid; i < DEC * HID; i += 256) {
        int k = i / HID, n = i - k * HID;
        W1Ts[n * W1T_STRIDE + k] = W1[i];
    }
    if (tid < HID) b1s[tid] = b1[tid];
    for (int i = tid; i < HID * 3; i += 256) W2s[i] = W2[i];
    if (tid < 3)   b2s[tid] = b2[tid];
    if (tid < 192) Woffs[tid] = Woff[tid];
    if (tid < OFFD) boffs[tid] = boff[tid];

    const int wave = tid >> 5;
    const int lane = tid & 31;
    const int m    = lane & 15;   // point index within wave tile
    const int half = lane >> 4;   // lane half (K-offset selector)

    float* featw = feats + wave * 16 * FEAT_STRIDE;

    const long pbase = (long)blockIdx.x * 128 + wave * 16;
    const long pg    = pbase + m;
    const int  p     = (pg < npts) ? (int)pg : (npts - 1);  // clamp: keep EXEC all-ones

    __syncthreads();

    // ---- Stage features: lane pair (m, half) builds one 192-float row ----
    float* frow = featw + m * FEAT_STRIDE;
    {
        const int x = coords[p * 3 + 0];
        const int y = coords[p * 3 + 1];
        const int z = coords[p * 3 + 2];
        const int vbase = z * 4096 + y * 64 + x;
        if (half == 0) {
            const float* pf = pfeat + (long)p * CPT;
            #pragma unroll
            for (int i = 0; i < CPT; ++i) frow[i] = pf[i];
            for (int c = 0; c < 64; ++c)
                frow[CPT + c] = vox[c * GRID3 + vbase];
        } else {
            for (int c = 64; c < CVOX; ++c)
                frow[CPT + c] = vox[c * GRID3 + vbase];
            const float px = poff[p * 3 + 0];
            const float py = poff[p * 3 + 1];
            const float pz = poff[p * 3 + 2];
            #pragma unroll
            for (int j = 0; j < OFFD; ++j)
                frow[CPT + CVOX + j] =
                    boffs[j] + px * Woffs[j] + py * Woffs[64 + j] + pz * Woffs[128 + j];
        }
    }
    __syncthreads();

    // ---- A fragments (16x4 f32 WMMA layout) ----
    // lane L: M = L&15; VGPR0 holds K = k0 + 2*half, VGPR1 holds K = k0 + 2*half + 1
    // A pair is two consecutive floats in the feat row -> single ds_load_b64.
    const int koff = half * 2;
    const float* abase = frow + koff;
    v2f areg[48];
    #pragma unroll
    for (int ks = 0; ks < 48; ++ks)
        areg[ks] = *(const v2f*)(abase + ks * 4);

    // ---- First GEMM: feat(16x192) @ W1(192x96) via V_WMMA_F32_16X16X4_F32 ----
    float* hld = featw;  // reuse feat region for h (stride H_STRIDE); wave is lockstep, safe
    #pragma unroll
    for (int t = 0; t < 6; ++t) {
        v8f acc = {0.f, 0.f, 0.f, 0.f, 0.f, 0.f, 0.f, 0.f};
        const int n = t * 16 + m;
        // B 4x16 layout: lanes 0-15 rows K=0,1 ; lanes 16-31 rows K=2,3.
        // With W1 transposed in LDS, the pair is consecutive -> single ds_load_b64.
        const float* bbase = W1Ts + n * W1T_STRIDE + koff;
        #pragma unroll
        for (int ks = 0; ks < 48; ++ks) {
            v2f b = *(const v2f*)(bbase + ks * 4);
            acc = __builtin_amdgcn_wmma_f32_16x16x4_f32(
                /*neg_a=*/false, areg[ks], /*neg_b=*/false, b,
                /*c_mod=*/(short)0, acc, /*reuse_a=*/false, /*reuse_b=*/false);
        }
        // C/D layout: VGPR j -> M = j + 8*half ; N = (lane&15) + 16*t
        #pragma unroll
        for (int j = 0; j < 8; ++j) {
            const int row = half * 8 + j;
            float v = acc[j] + b1s[n];
            v = 0.5f * v * (1.0f + erff(v * 0.70710678118654752f));  // exact gelu
            hld[row * H_STRIDE + n] = v;
        }
    }
    __syncthreads();

    // ---- Second GEMM: h(16x96) @ W2(96x3) + b2, split across lane halves ----
    float o0 = 0.f, o1 = 0.f, o2 = 0.f;
    const float* hr = hld + m * H_STRIDE + half * 48;
    #pragma unroll
    for (int k2 = 0; k2 < 48; ++k2) {
        const float hv = hr[k2];
        const int kk = half * 48 + k2;
        o0 += hv * W2s[kk * 3 + 0];
        o1 += hv * W2s[kk * 3 + 1];
        o2 += hv * W2s[kk * 3 + 2];
    }
    o0 += __shfl_xor(o0, 16);
    o1 += __shfl_xor(o1, 16);
    o2 += __shfl_xor(o2, 16);

    if (half == 0 && pg < npts) {
        out[pg * 3 + 0] = o0 + b2s[0];
        out[pg * 3 + 1] = o1 + b2s[1];
        out[pg * 3 + 2] = o2 + b2s[2];
    }
}

extern "C" void kernel_launch(void* const* d_in, const int* in_sizes, int n_in,
                              void* d_out, int out_size, void* d_ws, size_t ws_size,
                              hipStream_t stream) {
    const float* vox    = (const float*)d_in[0];
    const int*   coords = (const int*)  d_in[1];
    const float* pfeat  = (const float*)d_in[2];
    const float* poff   = (const float*)d_in[3];
    const float* Woff   = (const float*)d_in[4];
    const float* boff   = (const float*)d_in[5];
    const float* W1     = (const float*)d_in[6];
    const float* b1     = (const float*)d_in[7];
    const float* W2     = (const float*)d_in[8];
    const float* b2     = (const float*)d_in[9];
    float* out = (float*)d_out;

    const int npts = in_sizes[3] / 3;             // point_offsets is (N,3)
    const int nblocks = (npts + 127) / 128;       // 128 points per 256-thread block
    const size_t smem_bytes = (size_t)SMEM_FLOATS * sizeof(float);

    mambaflow_head_kernel<<<nblocks, 256, smem_bytes, stream>>>(
        vox, coords, pfeat, poff, Woff, boff, W1, b1, W2, b2, out, npts);
}